// ModelNew_25056839205024
// MI455X (gfx1250) — compile-verified
//
#include <hip/hip_runtime.h>
#include <hip/hip_bf16.h>
#include <stddef.h>

// Problem constants (from reference setup_inputs)
#define BB     4
#define SS     8192
#define HH     32
#define PP     64
#define NN     16
#define CHUNK  64
#define NSEG   8
#define SEG_LEN    (SS / NSEG)          // 1024
#define SEG_CHUNKS (SEG_LEN / CHUNK)    // 16
#define OUTSZ  (BB * HH * PP * NN)      // 131072
#define XPAD   16                        // LDS row pad (floats) -> conflict-free half-wave reads

typedef __attribute__((ext_vector_type(2))) float v2f;
typedef __attribute__((ext_vector_type(8))) float v8f;
typedef __attribute__((ext_vector_type(4))) int   v4i;

#define AS1 __attribute__((address_space(1)))
#define AS3 __attribute__((address_space(3)))

#if __has_builtin(__builtin_amdgcn_global_load_async_to_lds_b128)
#define HAS_ASYNC_LDS 1
#else
#define HAS_ASYNC_LDS 0
#endif

__device__ __forceinline__ void wait_async0() {
#if __has_builtin(__builtin_amdgcn_s_wait_asynccnt)
    __builtin_amdgcn_s_wait_asynccnt(0);
#else
    asm volatile("s_wait_asynccnt 0x0" ::: "memory");
#endif
}

// Copy one 16B granule global -> LDS (async if available).
// Builtin signature (from hipcc diagnostic): (v4i AS1*, v4i AS3*, imm i32, imm i32)
__device__ __forceinline__ void copy16(const float* src, float* dst) {
#if HAS_ASYNC_LDS
    __builtin_amdgcn_global_load_async_to_lds_b128((AS1 v4i*)src, (AS3 v4i*)dst,
                                                   /*imm_offset=*/0, /*cpol=*/0);
#else
    *(float4*)dst = *(const float4*)src;
#endif
}

// One workgroup per (b, h, seg). 128 threads = 4 waves; wave w owns p-tile [16w, 16w+16).
// Computes  Dseg[p,n] = sum_{s in seg} exp(Cum_segend - Cum_s) * X[s,p] * B[s,n]
// then scales by exp(sum of A after segment) and stores a partial tile to d_ws.
__global__ __launch_bounds__(128)
void ssd_state_partial_kernel(const float* __restrict__ X,
                              const float* __restrict__ A,
                              const float* __restrict__ Bm,
                              float* __restrict__ partial) {
    __shared__ float sX[2][CHUNK][PP + XPAD];  // 2 x 64 x 80 f32 = 40 KB (double-buffered X tile)
    __shared__ float sB[2][CHUNK][NN];         // 2 x 64 x 16 f32 = 8 KB  (double-buffered B tile)
    __shared__ float s_a[CHUNK];               // A-cumsum scratch
    __shared__ float s_w[CHUNK];               // per-step weights exp(ct - cum)
    __shared__ float s_red[128];               // tail-sum reduction

    const int t   = threadIdx.x;
    const int wg  = blockIdx.x;
    const int seg = wg % NSEG;
    const int h   = (wg / NSEG) % HH;
    const int b   = wg / (NSEG * HH);

    const int lane  = t & 31;
    const int wv    = t >> 5;            // wave id -> p-tile
    const int m     = lane & 15;         // M (for A op) / N (for B op) index
    const int khalf = (lane >> 4) << 1;  // 0 for lanes 0-15, 2 for lanes 16-31
    const int p0    = wv * 16;

    // Base pointers for this (b, h)
    const float* Xp = X  + (size_t)b * SS * HH * PP + (size_t)h * PP; // + s*H*P + p
    const float* Bp = Bm + (size_t)b * SS * HH * NN + (size_t)h * NN; // + s*H*N + n
    const float* Ap = A  + (size_t)b * SS * HH      + (size_t)h;      // + s*H
    const size_t XS = (size_t)HH * PP;   // 2048 floats per s-step
    const size_t BS = (size_t)HH * NN;   // 512
    const size_t AS = (size_t)HH;        // 32

    // ---- tail = sum of A over all steps AFTER this segment (decay seg-end -> seq-end) ----
    {
        float ts = 0.f;
        for (int s = (seg + 1) * SEG_LEN + t; s < SS; s += 128)
            ts += Ap[(size_t)s * AS];
        s_red[t] = ts;
        __syncthreads();
        for (int off = 64; off > 0; off >>= 1) {
            if (t < off) s_red[t] += s_red[t + off];
            __syncthreads();
        }
    }
    const float tail = s_red[0];
    __syncthreads();

    const int seg_s0 = seg * SEG_LEN;

    // Stage one 64-step chunk (X tile + B tile) into LDS buffer `buf`.
    auto stage = [&](int buf, int ci) {
        const int s_base = seg_s0 + ci * CHUNK;
        // X tile: 64 rows x 64 floats = 1024 x 16B granules; 8 per thread.
        #pragma unroll
        for (int i = 0; i < 8; ++i) {
            const int g   = t + 128 * i;
            const int s   = g >> 4;        // row 0..63
            const int c16 = g & 15;        // 16B granule within row
            copy16(Xp + (size_t)(s_base + s) * XS + c16 * 4, &sX[buf][s][c16 * 4]);
        }
        // B tile: 64 rows x 16 floats = 256 x 16B granules; 2 per thread.
        #pragma unroll
        for (int i = 0; i < 2; ++i) {
            const int g  = t + 128 * i;
            const int s  = g >> 2;         // row 0..63
            const int c4 = g & 3;          // 16B granule within row
            copy16(Bp + (size_t)(s_base + s) * BS + c4 * 4, &sB[buf][s][c4 * 4]);
        }
    };

    v8f acc = {};  // 16x16 f32 accumulator tile (8 VGPRs)

    stage(0, 0);   // prime the pipeline

    for (int ci = 0; ci < SEG_CHUNKS; ++ci) {
        const int cur = ci & 1;

        wait_async0();
        __syncthreads();   // chunk ci staged & visible; previous buffer free for reuse

        if (ci + 1 < SEG_CHUNKS) stage(cur ^ 1, ci + 1);  // overlap next chunk's HBM traffic

        // ---- A-cumsum scan for this chunk ----
        const int s_base = seg_s0 + ci * CHUNK;
        if (t < CHUNK) s_a[t] = Ap[(size_t)(s_base + t) * AS];
        __syncthreads();
        #pragma unroll
        for (int off = 1; off < CHUNK; off <<= 1) {
            float v = 0.f;
            if (t < CHUNK && t >= off) v = s_a[t - off];
            __syncthreads();
            if (t < CHUNK) s_a[t] += v;
            __syncthreads();
        }
        const float ct = s_a[CHUNK - 1];              // chunk total (<= 0)
        if (t < CHUNK) s_w[t] = __expf(ct - s_a[t]);  // weight to chunk end, <= 1
        __syncthreads();

        // Fold weights into the B tile once (replaces 2 VALU mults per WMMA).
        #pragma unroll
        for (int i = 0; i < 8; ++i) {
            const int g = t + 128 * i;                // 1024 elements
            const int s = g >> 4;
            const int n = g & 15;
            sB[cur][s][n] *= s_w[s];
        }
        __syncthreads();

        // Decay previously accumulated state across this chunk
        const float cdec = __expf(ct);
        #pragma unroll
        for (int r = 0; r < 8; ++r) acc[r] *= cdec;

        // 16 WMMA steps of K=4 cover the 64-step chunk; operands from LDS.
        // A-op (16x4 f32): lanes 0-15 hold (M=m, K=0/1), lanes 16-31 (M=m, K=2/3).
        // B-op (4x16 f32): lanes 0-15 hold (K=0/1, N=m), lanes 16-31 (K=2/3, N=m).
        #pragma unroll 8
        for (int k4 = 0; k4 < CHUNK / 4; ++k4) {
            const int so = (k4 << 2) + khalf;
            v2f av, bv;
            av.x = sX[cur][so][p0 + m];
            av.y = sX[cur][so + 1][p0 + m];
            bv.x = sB[cur][so][m];
            bv.y = sB[cur][so + 1][m];
            acc = __builtin_amdgcn_wmma_f32_16x16x4_f32(
                /*neg_a=*/false, av, /*neg_b=*/false, bv,
                /*c_mod=*/(short)0, acc, /*reuse_a=*/false, /*reuse_b=*/false);
        }
        // No trailing barrier needed: next iteration's top barrier orders buffer reuse.
    }

    // Apply cross-segment decay and store partial tile.
    // C/D layout: VGPR r, lanes 0-15 -> (M=r, N=lane); lanes 16-31 -> (M=8+r, N=lane-16).
    const float fin = __expf(tail);
    float* outp = partial + (size_t)seg * OUTSZ + ((size_t)(b * HH + h) * PP) * NN;
    const int mrow = p0 + ((lane >> 4) << 3);
    #pragma unroll
    for (int r = 0; r < 8; ++r) {
        outp[(size_t)(mrow + r) * NN + m] = acc[r] * fin;
    }
}

__global__ __launch_bounds__(256)
void ssd_state_reduce_kernel(const float* __restrict__ partial,
                             float* __restrict__ out) {
    const int idx = blockIdx.x * blockDim.x + threadIdx.x;
    if (idx >= OUTSZ) return;
    float s = 0.f;
    #pragma unroll
    for (int g = 0; g < NSEG; ++g)
        s += partial[(size_t)g * OUTSZ + idx];
    out[idx] = s;
}

extern "C" void kernel_launch(void* const* d_in, const int* in_sizes, int n_in,
                              void* d_out, int out_size, void* d_ws, size_t ws_size,
                              hipStream_t stream) {
    const float* X  = (const float*)d_in[0];   // [B,S,H,P]
    const float* A  = (const float*)d_in[1];   // [B,S,H]
    const float* Bm = (const float*)d_in[2];   // [B,S,H,N]
    float* out      = (float*)d_out;           // [B,H,P,N]
    float* part     = (float*)d_ws;            // [NSEG,B,H,P,N] partials (4 MB)

    dim3 grid1(BB * HH * NSEG);
    dim3 blk1(128);
    ssd_state_partial_kernel<<<grid1, blk1, 0, stream>>>(X, A, Bm, part);

    dim3 grid2((OUTSZ + 255) / 256);
    dim3 blk2(256);
    ssd_state_reduce_kernel<<<grid2, blk2, 0, stream>>>(part, out);
}